// CausalQueue_60773787238608
// MI455X (gfx1250) — compile-verified
//
#include <hip/hip_runtime.h>
#include <hip/hip_bf16.h>

// CausalQueue: out[b, 0:C]   = full ? buffer[head, b, :] : 0
//              out[b, C:2C]  = x[b, :]
// D=128, B=2048, C=512, all float32. Pure streaming copy (0 FLOPs, ~16.8 MB
// of HBM traffic -> ~0.72 us at 23.3 TB/s). Implemented with the CDNA5 async
// global<->LDS DMA path (ASYNCcnt) using 128-bit transfers.

#define QD   128
#define QB   2048
#define QC   512

#define ROW4   ((2 * QC) / 4)   // 256 float4 per output row
#define HALF4  (QC / 4)         // 128 float4 per half-row
#define TOTAL4 (QB * ROW4)      // 524288 float4 total
#define TPB    256
#define KPT    4                 // float4 tiles per thread
#define NBLK   (TOTAL4 / (TPB * KPT))  // 512 blocks

__global__ __launch_bounds__(TPB)
void causal_queue_concat_kernel(const float4* __restrict__ x4,
                                const float4* __restrict__ buf4,
                                const int*    __restrict__ size_p,
                                const int*    __restrict__ head_p,
                                float4*       __restrict__ out4)
{
    // 16 KB LDS staging: one disjoint slot per (thread, k); no reuse, so a
    // single s_wait_asynccnt between the load phase and the store phase is
    // the only synchronization required. Store completion is covered by the
    // implicit wait-idle at S_ENDPGM.
    __shared__ float4 stage[TPB * KPT];

    const bool full = (size_p[0] == QD);
    const int  head = head_p[0];
    const float4* __restrict__ past4 =
        buf4 + (size_t)head * (size_t)(QB * HALF4);

    // Low 32 bits of a generic LDS pointer are the in-LDS byte offset
    // (aperture occupies bits [63:32]).
    const unsigned lds_base = (unsigned)(uintptr_t)(&stage[0]);

    const int base = (int)blockIdx.x * (TPB * KPT);

    int  idx[KPT];
    bool via_async[KPT];

#pragma unroll
    for (int k = 0; k < KPT; ++k) {
        const int i = base + k * TPB + (int)threadIdx.x;
        idx[k] = i;
        const int  j    = i & (ROW4 - 1);   // column (float4 units) in out row
        const int  b    = i >> 8;           // row; ROW4 == 256
        const bool past = (j < HALF4);      // wave-uniform: HALF4 % 32 == 0

        if (past && !full) {
            via_async[k] = false;           // cold start: zeros, no load
        } else {
            via_async[k] = true;
            const float4* src = past ? (past4 + (size_t)b * HALF4 + j)
                                     : (x4    + (size_t)b * HALF4 + (j - HALF4));
            const unsigned long long ga = (unsigned long long)(uintptr_t)src;
            const unsigned lds_addr =
                lds_base + (unsigned)((k * TPB + (int)threadIdx.x) * 16);
            // ASYNCcnt-tracked DMA: global -> LDS, 128 bits per lane.
            asm volatile("global_load_async_to_lds_b128 %0, %1, off"
                         :
                         : "v"(lds_addr), "v"(ga)
                         : "memory");
        }
    }

    // Drain this wave's async loads before the LDS slots are read back.
    asm volatile("s_wait_asynccnt 0" ::: "memory");

#pragma unroll
    for (int k = 0; k < KPT; ++k) {
        float4* dst = out4 + idx[k];
        if (via_async[k]) {
            const unsigned long long ga = (unsigned long long)(uintptr_t)dst;
            const unsigned lds_addr =
                lds_base + (unsigned)((k * TPB + (int)threadIdx.x) * 16);
            // ASYNCcnt-tracked DMA: LDS -> global, 128 bits per lane.
            asm volatile("global_store_async_from_lds_b128 %0, %1, off"
                         :
                         : "v"(ga), "v"(lds_addr)
                         : "memory");
        } else {
            const float4 z = {0.0f, 0.0f, 0.0f, 0.0f};
            __builtin_nontemporal_store(z.x, &dst->x);
            __builtin_nontemporal_store(z.y, &dst->y);
            __builtin_nontemporal_store(z.z, &dst->z);
            __builtin_nontemporal_store(z.w, &dst->w);
        }
    }
    // s_endpgm performs an implicit wait-idle -> outstanding async stores drain.
}

extern "C" void kernel_launch(void* const* d_in, const int* in_sizes, int n_in,
                              void* d_out, int out_size, void* d_ws, size_t ws_size,
                              hipStream_t stream) {
    (void)in_sizes; (void)n_in; (void)d_ws; (void)ws_size; (void)out_size;

    const float4* x4    = (const float4*)d_in[0];  // [B, C] f32
    const float4* buf4  = (const float4*)d_in[1];  // [D, B, C] f32
    const int*    size_p = (const int*)d_in[2];    // scalar
    const int*    head_p = (const int*)d_in[3];    // scalar
    float4*       out4  = (float4*)d_out;          // [B, 2C] f32

    causal_queue_concat_kernel<<<NBLK, TPB, 0, stream>>>(x4, buf4, size_p,
                                                         head_p, out4);
}